// AttentionNet_60421599920604
// MI455X (gfx1250) — compile-verified
//
#include <hip/hip_runtime.h>

// Problem constants (reference: L=512, B=4, E=256)
constexpr int Lc = 512;
constexpr int Bc = 4;
constexpr int Ec = 256;
constexpr int COLS = Bc * Ec;    // 1024 scan columns
constexpr int NCH = 16;          // cumsum chunks along L
constexpr int CLEN = Lc / NCH;   // 32

typedef __attribute__((ext_vector_type(2))) float v2f;
typedef __attribute__((ext_vector_type(8))) float v8f;

// Single-TRANS tanh (v_tanh_f32 on gfx1250); exp2-based fallback.
#if __has_builtin(__builtin_amdgcn_tanhf)
__device__ __forceinline__ float fast_tanh(float x) { return __builtin_amdgcn_tanhf(x); }
#else
__device__ __forceinline__ float fast_tanh(float x) {
    // tanh(x) = 1 - 2/(1 + e^{2x}),  e^{2x} = 2^{2*log2(e)*x}
    return 1.0f - 2.0f * __builtin_amdgcn_rcpf(1.0f + __builtin_amdgcn_exp2f(2.885390082f * x));
}
#endif

// ---------------- Phase 1: S = cumsum_l(emb), 3-step chunked scan ----------------
__global__ void scan_chunksum(const float* __restrict__ emb, float* __restrict__ Csum) {
    const int idx = blockIdx.x * blockDim.x + threadIdx.x;  // chunk*COLS + col
    if (idx >= NCH * COLS) return;
    const int col = idx % COLS, ch = idx / COLS;
    const float* p = emb + (size_t)ch * CLEN * COLS + col;
    float s = 0.0f;
    #pragma unroll 8
    for (int l = 0; l < CLEN; ++l) s += p[l * COLS];
    Csum[ch * COLS + col] = s;
}

__global__ void scan_offsets(const float* __restrict__ Csum, float* __restrict__ Coff) {
    const int col = blockIdx.x * blockDim.x + threadIdx.x;
    if (col >= COLS) return;
    float run = 0.0f;
    #pragma unroll
    for (int ch = 0; ch < NCH; ++ch) {
        Coff[ch * COLS + col] = run;
        run += Csum[ch * COLS + col];
    }
}

__global__ void scan_expand(const float* __restrict__ emb, const float* __restrict__ Coff,
                            float* __restrict__ S) {
    const int idx = blockIdx.x * blockDim.x + threadIdx.x;
    if (idx >= NCH * COLS) return;
    const int col = idx % COLS, ch = idx / COLS;
    const float* p = emb + (size_t)ch * CLEN * COLS + col;
    float* q = S + (size_t)ch * CLEN * COLS + col;
    float run = Coff[ch * COLS + col];
    #pragma unroll 8
    for (int l = 0; l < CLEN; ++l) { run += p[l * COLS]; q[l * COLS] = run; }
}

// ---------------- Phase 2: U' = 0.5*(X@W1^T) ; V' = 0.5*(X@W2^T + S@W3^T + b3) ------------
// Wave computes 4 M-tiles x 1 N-tile (64x16): 11 b64 loads per 12 WMMAs per k-step.
__global__ void __launch_bounds__(128) uv_gemm_kernel(
        const float* __restrict__ X, const float* __restrict__ S,
        const float* __restrict__ w1, const float* __restrict__ w2,
        const float* __restrict__ w3, const float* __restrict__ b3,
        float* __restrict__ U, float* __restrict__ V) {
    const int wave = blockIdx.x * (blockDim.x >> 5) + (threadIdx.x >> 5);  // 0..511
    const int lane = threadIdx.x & 31;
    const int lr = lane & 15, hf = lane >> 4;
    constexpr int N_TILES = Ec / 16;          // 16
    constexpr int M4_TILES = (Lc * Bc) / 64;  // 32
    const int nt = wave % N_TILES;
    const int mt4 = wave / N_TILES;
    if (mt4 >= M4_TILES) return;              // wave-uniform guard
    const int n0 = nt * 16;

    v8f accU[4] = {{}, {}, {}, {}};
    v8f accV[4] = {{}, {}, {}, {}};
    const float* p1 = w1 + (n0 + lr) * Ec + 2 * hf;
    const float* p2 = w2 + (n0 + lr) * Ec + 2 * hf;
    const float* p3 = w3 + (n0 + lr) * Ec + 2 * hf;
    const float* px0 = X + (mt4 * 64 + lr) * Ec + 2 * hf;
    const float* ps0 = S + (mt4 * 64 + lr) * Ec + 2 * hf;

    for (int k0 = 0; k0 < Ec; k0 += 4) {
        v2f b1 = { p1[k0], p1[k0 + 1] };
        v2f b2 = { p2[k0], p2[k0 + 1] };
        v2f b3v = { p3[k0], p3[k0 + 1] };
        #pragma unroll
        for (int s4 = 0; s4 < 4; ++s4) {
            const float* px = px0 + s4 * 16 * Ec;
            const float* ps = ps0 + s4 * 16 * Ec;
            v2f ax = { px[k0], px[k0 + 1] };
            v2f as = { ps[k0], ps[k0 + 1] };
            accU[s4] = __builtin_amdgcn_wmma_f32_16x16x4_f32(false, ax, false, b1, (short)0, accU[s4], false, false);
            accV[s4] = __builtin_amdgcn_wmma_f32_16x16x4_f32(false, ax, false, b2, (short)0, accV[s4], false, false);
            accV[s4] = __builtin_amdgcn_wmma_f32_16x16x4_f32(false, as, false, b3v, (short)0, accV[s4], false, false);
        }
    }

    const float bias = b3[n0 + lr];
    #pragma unroll
    for (int s4 = 0; s4 < 4; ++s4) {
        #pragma unroll
        for (int j = 0; j < 8; ++j) {
            const int m = mt4 * 64 + s4 * 16 + j + 8 * hf;
            U[m * Ec + n0 + lr] = 0.5f * accU[s4][j];                 // U' = u/2
            V[m * Ec + n0 + lr] = 0.5f * (accV[s4][j] + bias);        // V' = (q+c+b3)/2
        }
    }
}

// ---------------- Phase 3 (dominant): scores[b,t,i] = sum_e hw0[e]*(1+tanh(U'+V')), i<=t ---
// add + v_tanh_f32 + fma per element. Block = (t,b); wave per i; lanes split e.
// NOTE: scores stored as (b, t, i) so phase 4's K-dimension (i) is contiguous.
__global__ void __launch_bounds__(256) scores_kernel(
        const float* __restrict__ U, const float* __restrict__ V,
        const float* __restrict__ w0, float* __restrict__ scores) {
    const int t = blockIdx.x;
    const int b = blockIdx.y;
    const int wid = threadIdx.x >> 5;
    const int lane = threadIdx.x & 31;

    const float4* w0p = (const float4*)(w0 + lane * 8);
    float4 ha = w0p[0], hb = w0p[1];
    ha.x *= 0.5f; ha.y *= 0.5f; ha.z *= 0.5f; ha.w *= 0.5f;
    hb.x *= 0.5f; hb.y *= 0.5f; hb.z *= 0.5f; hb.w *= 0.5f;
    // lane-partial of sum_e w0[e]/2 (the "+1" term of 1+tanh)
    const float s0 = ha.x + ha.y + ha.z + ha.w + hb.x + hb.y + hb.z + hb.w;

    const float4* vp = (const float4*)(V + (t * Bc + b) * Ec + lane * 8);
    const float4 va = vp[0], vb = vp[1];

    const int i_max = ((t >> 4) + 1) << 4;  // round t+1 up to tile of 16
    for (int i = wid; i < i_max; i += 8) {
        float s = 0.0f;
        if (i <= t) {  // wave-uniform branch
            const float4* up = (const float4*)(U + (i * Bc + b) * Ec + lane * 8);
            const float4 ua = up[0], ub = up[1];
            s = s0;
            s = fmaf(ha.x, fast_tanh(ua.x + va.x), s);
            s = fmaf(ha.y, fast_tanh(ua.y + va.y), s);
            s = fmaf(ha.z, fast_tanh(ua.z + va.z), s);
            s = fmaf(ha.w, fast_tanh(ua.w + va.w), s);
            s = fmaf(hb.x, fast_tanh(ub.x + vb.x), s);
            s = fmaf(hb.y, fast_tanh(ub.y + vb.y), s);
            s = fmaf(hb.z, fast_tanh(ub.z + vb.z), s);
            s = fmaf(hb.w, fast_tanh(ub.w + vb.w), s);
            #pragma unroll
            for (int off = 16; off; off >>= 1) s += __shfl_xor(s, off, 32);
        }
        if (lane == 0) scores[((size_t)b * Lc + t) * Lc + i] = s;  // zero-fills fringe
    }
}

// ---------------- Phase 4: out[t,b,:] = sum_{i<=t} scores[b,t,i] * emb[i,b,:] ---------------
// Wave computes 4 consecutive t-tiles x 1 e-tile, sharing the B (emb) loads.
// A loads are contiguous in i -> single b64 per operand. Causal staircase k-loops.
__global__ void __launch_bounds__(128) out_gemm_kernel(
        const float* __restrict__ scores, const float* __restrict__ emb,
        float* __restrict__ out) {
    const int wave = blockIdx.x * (blockDim.x >> 5) + (threadIdx.x >> 5);  // 0..511
    const int lane = threadIdx.x & 31;
    const int lr = lane & 15, hf = lane >> 4;
    constexpr int N_TILES = Ec / 16;   // 16
    constexpr int T4_TILES = Lc / 64;  // 8
    const int nt  = wave % N_TILES;
    const int tt4 = (wave / N_TILES) % T4_TILES;
    const int b   = wave / (N_TILES * T4_TILES);
    if (b >= Bc) return;               // wave-uniform guard
    const int t0 = tt4 * 64, e0 = nt * 16;

    v8f acc[4] = {{}, {}, {}, {}};
    const float* aBase = scores + ((size_t)b * Lc + t0 + lr) * Lc;  // row t0+lr, i contiguous
    const float* bBase = emb + (size_t)b * Ec + e0 + lr;            // B[i][e]: i-stride B*E

    int k0 = 0;
    // bulk: all 4 sub-tiles active for k0 < t0+16 (sub 0 ends at t0+16)
    #define OUT_STEP(SLO)                                                        \
        {                                                                        \
            const int k = k0 + 2 * hf;                                           \
            const float* pb = bBase + (size_t)k * (Bc * Ec);                     \
            v2f bv = { pb[0], pb[Bc * Ec] };                                     \
            _Pragma("unroll")                                                    \
            for (int s = (SLO); s < 4; ++s) {                                    \
                const float* pa = aBase + (size_t)s * 16 * Lc + k;               \
                v2f av = { pa[0], pa[1] };                                       \
                acc[s] = __builtin_amdgcn_wmma_f32_16x16x4_f32(                  \
                    false, av, false, bv, (short)0, acc[s], false, false);       \
            }                                                                    \
        }
    for (; k0 < t0 + 16; k0 += 4) OUT_STEP(0)   // subs 0..3
    for (; k0 < t0 + 32; k0 += 4) OUT_STEP(1)   // subs 1..3
    for (; k0 < t0 + 48; k0 += 4) OUT_STEP(2)   // subs 2..3
    for (; k0 < t0 + 64; k0 += 4) OUT_STEP(3)   // sub 3
    #undef OUT_STEP

    #pragma unroll
    for (int s = 0; s < 4; ++s) {
        #pragma unroll
        for (int j = 0; j < 8; ++j) {
            const int t = t0 + s * 16 + j + 8 * hf;
            out[(t * Bc + b) * Ec + e0 + lr] = acc[s][j];
        }
    }
}

extern "C" void kernel_launch(void* const* d_in, const int* in_sizes, int n_in,
                              void* d_out, int out_size, void* d_ws, size_t ws_size,
                              hipStream_t stream) {
    const float* emb = (const float*)d_in[0];
    const float* w1  = (const float*)d_in[1];
    const float* w2  = (const float*)d_in[2];
    const float* w3  = (const float*)d_in[3];
    const float* b3  = (const float*)d_in[4];
    const float* w0  = (const float*)d_in[5];
    float* out = (float*)d_out;

    // Workspace (floats): S | U | V | scores | Csum | Coff  (~10.6 MB)
    float* S      = (float*)d_ws;
    float* U      = S + (size_t)Lc * Bc * Ec;
    float* V      = U + (size_t)Lc * Bc * Ec;
    float* scores = V + (size_t)Lc * Bc * Ec;
    float* Csum   = scores + (size_t)Lc * Lc * Bc;
    float* Coff   = Csum + (size_t)NCH * COLS;

    // Phase 1: chunked parallel cumsum (dep chains of 32/16/32 instead of 512)
    scan_chunksum<<<dim3((NCH * COLS + 255) / 256), dim3(256), 0, stream>>>(emb, Csum);
    scan_offsets<<<dim3((COLS + 255) / 256), dim3(256), 0, stream>>>(Csum, Coff);
    scan_expand<<<dim3((NCH * COLS + 255) / 256), dim3(256), 0, stream>>>(emb, Coff, S);

    // Phase 2: U'/V' projections. 32x16 = 512 waves, 4 waves per 128-thread block.
    uv_gemm_kernel<<<dim3(512 / 4), dim3(128), 0, stream>>>(emb, S, w1, w2, w3, b3, U, V);

    // Phase 3: fused tanh-sigmoid + dot(w0) with causal mask. One block per (t, b).
    scores_kernel<<<dim3(Lc, Bc), dim3(256), 0, stream>>>(U, V, w0, scores);

    // Phase 4: causal batched GEMM scores @ emb. 4 t-tiles per wave, 512 waves.
    out_gemm_kernel<<<dim3(512 / 4), dim3(128), 0, stream>>>(scores, emb, out);
}